// DCModule_25451976196444
// MI455X (gfx1250) — compile-verified
//
#include <hip/hip_runtime.h>
#include <hip/hip_bf16.h>

// Problem geometry
#define H      4096
#define W      4096
#define NWIN   2047          // windows per dimension (stride 2, win 3)
// Tiling: each 256-thread WG handles TH x TW output 2x2-blocks (== windows)
#define TW     32
#define TH     8
#define LROWS  17            // 2*TH + 1 input rows needed
#define NCHUNK 17            // ceil((2*TW+1)/4) float4 chunks per row
#define LCOLS  (NCHUNK * 4)  // 68 floats LDS row stride (16B multiple)

// ---- CDNA5 async global->LDS path (existence proven by round-1 diagnostics) ----
#if defined(__has_builtin)
#if __has_builtin(__builtin_amdgcn_global_load_async_to_lds_b128)
#define HAVE_ASYNC_B128 1
#endif
#if __has_builtin(__builtin_amdgcn_s_wait_asynccnt)
#define HAVE_WAIT_ASYNC 1
#endif
#endif

// 16-byte int vector with explicit address spaces (matches builtin signature:
// param 0 = V4i in AS(1) global, param 1 = V4i in AS(3) LDS)
typedef int v4i __attribute__((ext_vector_type(4)));
typedef v4i __attribute__((address_space(1))) gv4i;
typedef v4i __attribute__((address_space(3))) lv4i;
typedef float v2f __attribute__((ext_vector_type(2)));

__device__ __forceinline__ void async_copy16(const float* g, float* l) {
#if HAVE_ASYNC_B128
  gv4i* gp = (gv4i*)g;   // explicit addrspacecast generic -> global
  lv4i* lp = (lv4i*)l;   // explicit addrspacecast generic -> LDS
  __builtin_amdgcn_global_load_async_to_lds_b128(gp, lp, /*offset=*/0, /*cpol=*/0);
#else
  *(float4*)l = *(const float4*)g;
#endif
}

__device__ __forceinline__ void wait_async_then_barrier() {
#if HAVE_ASYNC_B128
#if HAVE_WAIT_ASYNC
  __builtin_amdgcn_s_wait_asynccnt(0);
#else
  asm volatile("s_wait_asynccnt 0" ::: "memory");
#endif
#endif
  __syncthreads();
}

// min+max pooled sum for one window; first-index tie-break like jnp.arg{min,max}
__device__ __forceinline__ float pool_sum(const float a[9], const float c[9]) {
  float d0 = fabsf(a[0] - c[0]);
  float dmin = d0, dmax = d0, vmin = c[0], vmax = c[0];
#pragma unroll
  for (int k = 1; k < 9; ++k) {
    float d = fabsf(a[k] - c[k]);
    if (d < dmin) { dmin = d; vmin = c[k]; }   // strict <  -> first min wins
    if (d > dmax) { dmax = d; vmax = c[k]; }   // strict >  -> first max wins
  }
  return vmin + vmax;
}

__device__ __forceinline__ void load_win(const float (*tile)[LCOLS], int lr, int lc,
                                         float w[9]) {
#pragma unroll
  for (int r = 0; r < 3; ++r) {
    const float* p = &tile[lr + r][lc];     // 8B aligned (lc even) -> b64+b32 ds loads
    w[3 * r + 0] = p[0];
    w[3 * r + 1] = p[1];
    w[3 * r + 2] = p[2];
  }
}

__device__ __forceinline__ void emit_block(float* __restrict__ out,
                                           const float (*tile)[LCOLS],
                                           float v, int y0, int x0,
                                           int row0, int col0) {
  float o[2][2];
#pragma unroll
  for (int r = 0; r < 2; ++r)
#pragma unroll
    for (int c = 0; c < 2; ++c) {
      int y = y0 + r, x = x0 + c;
      bool cov = (y < H - 1) && (x < W - 1);
      // uncovered border = 2*comp, served from the LDS tile (indices proven in-range)
      o[r][c] = cov ? v : 2.0f * tile[y - row0][x - col0];
    }
  v2f s0 = {o[0][0], o[0][1]};
  v2f s1 = {o[1][0], o[1][1]};
  // outputs are write-once streams: NT stores keep the 192MB input set resident in L2
  __builtin_nontemporal_store(s0, (v2f*)&out[(size_t)y0 * W + x0]);
  __builtin_nontemporal_store(s1, (v2f*)&out[(size_t)(y0 + 1) * W + x0]);
}

__global__ __launch_bounds__(256) void tripool_kernel(
    const float* __restrict__ A, const float* __restrict__ P,
    const float* __restrict__ N, float* __restrict__ outP,
    float* __restrict__ outN) {
  // 16B alignment required: async b128 writes land directly in LDS
  __shared__ __align__(16) float lds[3][LROWS][LCOLS];  // anchor / pos / neg tiles

  const int tid  = threadIdx.x;
  const int bx0  = blockIdx.x * TW;
  const int by0  = blockIdx.y * TH;
  const int row0 = 2 * by0;
  const int col0 = 2 * bx0;

  // ---- stage: 3 matrices x 17 rows x 17 b128 chunks = 867 async copies ----
  for (int i = tid; i < 3 * LROWS * NCHUNK; i += 256) {
    int m   = i / (LROWS * NCHUNK);
    int rem = i - m * (LROWS * NCHUNK);
    int r   = rem / NCHUNK;
    int c   = rem - r * NCHUNK;
    // clamp so every global access is in-range; clamped LDS slots are never read
    int grow = min(row0 + r, H - 1);
    int gcol = min(col0 + 4 * c, W - 4);
    const float* src = (m == 0) ? A : ((m == 1) ? P : N);
    async_copy16(src + (size_t)grow * W + gcol, &lds[m][r][4 * c]);
  }
  wait_async_then_barrier();

  // ---- compute: one window (= one 2x2 output block) per thread ----
  const int ly = tid >> 5;           // 0..7   (wave id)
  const int lx = tid & 31;           // 0..31  (lane)
  const int by = by0 + ly;
  const int bx = bx0 + lx;
  const int wi = min(by, NWIN - 1);  // window row (block 2047 reuses window 2046)
  const int wj = min(bx, NWIN - 1);
  const int lr = 2 * wi - row0;      // 0..14
  const int lc = 2 * wj - col0;      // 0..62 (even -> conflict-free banks)

  float a[9], p[9], n[9];
  load_win(lds[0], lr, lc, a);
  load_win(lds[1], lr, lc, p);
  load_win(lds[2], lr, lc, n);

  const float vp = pool_sum(a, p);
  const float vn = pool_sum(a, n);

  const int y0 = 2 * by, x0 = 2 * bx;
  emit_block(outP, lds[1], vp, y0, x0, row0, col0);
  emit_block(outN, lds[2], vn, y0, x0, row0, col0);
}

extern "C" void kernel_launch(void* const* d_in, const int* in_sizes, int n_in,
                              void* d_out, int out_size, void* d_ws, size_t ws_size,
                              hipStream_t stream) {
  const float* A = (const float*)d_in[0];
  const float* P = (const float*)d_in[1];
  const float* N = (const float*)d_in[2];
  float* outP = (float*)d_out;
  float* outN = outP + (size_t)H * W;

  dim3 grid(W / (2 * TW), H / (2 * TH));  // 64 x 256 workgroups
  tripool_kernel<<<grid, 256, 0, stream>>>(A, P, N, outP, outN);
}